// CompressedKVCache_16870631539365
// MI455X (gfx1250) — compile-verified
//
#include <hip/hip_runtime.h>
#include <stdint.h>

// CompressedKVCache: INT4 quantize round-trip (new tokens) + INT4 dequant
// (cached prefix), fused streaming kernel for MI455X (gfx1250, wave32).
//
// Bandwidth-bound: ~440 MB of traffic -> ~19 us at 23.3 TB/s. Index math is
// fully divide-free via a 3D grid (t, bh, n); the previous version wasted
// hundreds of VALU/SALU ops per wave on 64-bit division emulation.

typedef float v4f __attribute__((ext_vector_type(4)));

#define NB 2
#define NH 32
#define ND 128
#define WAVES_PER_BLOCK 8
#define BLOCK_THREADS (WAVES_PER_BLOCK * 32)

__global__ __launch_bounds__(BLOCK_THREADS) void
kvcache_q4_roundtrip_kernel(const float* __restrict__ k,
                            const float* __restrict__ v,
                            const uint8_t* __restrict__ k_cache,
                            const uint8_t* __restrict__ v_cache,
                            const float* __restrict__ k_scale,
                            const float* __restrict__ k_zero,
                            const float* __restrict__ v_scale,
                            const float* __restrict__ v_zero,
                            float* __restrict__ out,   // [2][NB*NH][end][ND]
                            int L, int start, int end, int max_seq) {
    const int wid  = threadIdx.x >> 5;            // wave id in block (wave32)
    const int lane = threadIdx.x & 31;

    const int t  = blockIdx.z;                    // 0 = k, 1 = v
    const int bh = blockIdx.y;                    // fused (b,h), 0..NB*NH-1
    const int n  = blockIdx.x * WAVES_PER_BLOCK + wid;  // sequence position
    if (n >= end) return;                         // wave-uniform exit

    const float*   src     = t ? v : k;
    const uint8_t* cache   = t ? v_cache : k_cache;
    const float*   scale_p = t ? v_scale : k_scale;
    const float*   zero_p  = t ? v_zero  : k_zero;

    float* dst = out + (((long)t * (NB * NH) + bh) * end + n) * ND + lane * 4;

    v4f o;
    if (n >= start) {
        // ---- new token: quantize -> dequantize round trip (matches
        // _quant_pack followed by _unpack_dequant bit-for-bit) ----
        const long m = (long)bh * L + (n - start);       // row in k/v input
        const v4f x = __builtin_nontemporal_load(
            (const v4f*)(src + m * ND + lane * 4));      // 512B/wave, TH=NT

        float lmin = fminf(fminf(x.x, x.y), fminf(x.z, x.w));
        float lmax = fmaxf(fmaxf(x.x, x.y), fmaxf(x.z, x.w));
        // wave32 butterfly reduction (all 32 lanes active)
        #pragma unroll
        for (int off = 16; off > 0; off >>= 1) {
            lmin = fminf(lmin, __shfl_xor(lmin, off, 32));
            lmax = fmaxf(lmax, __shfl_xor(lmax, off, 32));
        }
        const float scale = fmaxf((lmax - lmin) / 15.0f, 1e-8f);
        const float zero  = -lmin / scale;
        #pragma unroll
        for (int i = 0; i < 4; ++i) {
            float q = rintf(x[i] / scale + zero);        // round-half-even
            q = fminf(fmaxf(q, 0.0f), 15.0f);
            o[i] = (q - zero) * scale;
        }
    } else {
        // ---- cached prefix: unpack nibbles + dequantize ----
        const long cidx = (long)bh * max_seq + n;
        const uint16_t p = __builtin_nontemporal_load(
            (const uint16_t*)(cache + cidx * (ND / 2)) + lane);  // 64B/wave
        const float scale = scale_p[cidx];
        const float zero  = zero_p[cidx];
        o.x = ((float)( p        & 15) - zero) * scale;  // even <- low nibble
        o.y = ((float)((p >> 4)  & 15) - zero) * scale;  // odd  <- high nibble
        o.z = ((float)((p >> 8)  & 15) - zero) * scale;
        o.w = ((float)((p >> 12) & 15) - zero) * scale;
    }
    // streaming store: touched once, keep it out of L2's working set
    __builtin_nontemporal_store(o, (v4f*)dst);
}

extern "C" void kernel_launch(void* const* d_in, const int* in_sizes, int n_in,
                              void* d_out, int out_size, void* d_ws, size_t ws_size,
                              hipStream_t stream) {
    const float*   k       = (const float*)d_in[0];
    const float*   v       = (const float*)d_in[1];
    const uint8_t* k_cache = (const uint8_t*)d_in[2];
    const uint8_t* v_cache = (const uint8_t*)d_in[3];
    const float*   k_scale = (const float*)d_in[4];
    const float*   k_zero  = (const float*)d_in[5];
    const float*   v_scale = (const float*)d_in[6];
    const float*   v_zero  = (const float*)d_in[7];
    // d_in[8] = start_pos (device scalar) -- derived on host instead so the
    // launch is graph-capture safe and deterministic.

    const int L       = in_sizes[0] / (NB * NH * ND);            // 2048
    const int end_pos = out_size / (2 * NB * NH * ND);           // 4096
    const int start   = end_pos - L;                             // 2048
    const int max_seq = in_sizes[2] / (NB * NH * (ND / 2));      // 8192

    dim3 grid((end_pos + WAVES_PER_BLOCK - 1) / WAVES_PER_BLOCK, // 512
              NB * NH,                                           // 64
              2);                                                // k, v
    kvcache_q4_roundtrip_kernel<<<grid, BLOCK_THREADS, 0, stream>>>(
        k, v, k_cache, v_cache, k_scale, k_zero, v_scale, v_zero,
        (float*)d_out, L, start, end_pos, max_seq);
}